// GATLayer_74345883894233
// MI455X (gfx1250) — compile-verified
//
#include <hip/hip_runtime.h>

// ---------------------------------------------------------------------------
// GAT layer for MI455X (gfx1250, wave32).
//   N=30000 nodes, K=32 neighbors (== wave32!), D=128 features.
// Pipeline:
//   K1 gemm  : z = h @ Wfc^T           (V_WMMA_F32_16X16X4_F32, fp32 precise)
//   K2 dots  : s_src,t0,t1 (pack4), s_dst per node (wave dot-reductions)
//   K3 attn  : softmax over 32 lanes, row conv, col conv (gather from L2-
//              resident z table), per-block BN partial sums (deterministic)
//   K4 stats : single-block reduce -> mean / gamma*rsqrt(var+eps)
//   K5 final : affine + relu + concat -> out (N,159)
// ---------------------------------------------------------------------------

#define NN 30000
#define KK 32
#define DD 128

typedef __attribute__((ext_vector_type(2))) float v2f;
typedef __attribute__((ext_vector_type(8))) float v8f;

__device__ __forceinline__ float wave_sum(float v) {
#pragma unroll
  for (int m = 16; m >= 1; m >>= 1) v += __shfl_xor(v, m, 32);
  return v;
}
__device__ __forceinline__ float wave_max(float v) {
#pragma unroll
  for (int m = 16; m >= 1; m >>= 1) v = fmaxf(v, __shfl_xor(v, m, 32));
  return v;
}

// ---------------------------------------------------------------------------
// K1: z[m,n] = sum_k h[m,k] * Wfc[n,k].   One wave per 16x16 tile of z.
// Block = 256 threads = 8 waves covering one 16-row strip (8 n-tiles of 16).
// A (16x4 f32) lane layout (ISA 7.12.2): lanes 0-15 M=0..15, V0={K0|K2},
// V1={K1|K3} -> each lane holds the contiguous pair K = 2*half, 2*half+1.
// B (4x16) mirrored: lane n = lane&15, same K pairing. C/D: V0..7 = rows
// M=half*8+v, col = n0 + (lane&15).
// ---------------------------------------------------------------------------
__global__ __launch_bounds__(256) void gat_gemm_kernel(
    const float* __restrict__ h, const float* __restrict__ Wfc,
    float* __restrict__ z) {
  const int lane = threadIdx.x & 31;
  const int wave = threadIdx.x >> 5;
  const int m0 = blockIdx.x * 16;
  const int n0 = wave * 16;
  const int half = lane >> 4;
  const int l = lane & 15;

  const float* __restrict__ ha = h + (size_t)(m0 + l) * DD + 2 * half;
  const float* __restrict__ wb = Wfc + (size_t)(n0 + l) * DD + 2 * half;

  v8f c = {};
#pragma unroll 4
  for (int k0 = 0; k0 < DD; k0 += 4) {
    v2f a = {ha[k0], ha[k0 + 1]};
    v2f b = {wb[k0], wb[k0 + 1]};
    c = __builtin_amdgcn_wmma_f32_16x16x4_f32(
        /*neg_a=*/false, a, /*neg_b=*/false, b,
        /*c_mod=*/(short)0, c, /*reuse_a=*/false, /*reuse_b=*/false);
  }

  float* zd = z + (size_t)(m0 + half * 8) * DD + n0 + l;
#pragma unroll
  for (int v = 0; v < 8; ++v) zd[(size_t)v * DD] = c[v];
}

// ---------------------------------------------------------------------------
// K2: per-node scalars. One wave per node, 8 nodes per block.
//   pack4[j] = { z[j].a_src, z[j].wr0, z[j].wr1, 0 },  s_dst[j] = z[j].a_dst
// ---------------------------------------------------------------------------
__global__ __launch_bounds__(256) void gat_dots_kernel(
    const float* __restrict__ z, const float* __restrict__ Wa,
    const float* __restrict__ wr, float4* __restrict__ pack4,
    float* __restrict__ s_dst) {
  const int lane = threadIdx.x & 31;
  const int wave = threadIdx.x >> 5;
  const int n = blockIdx.x * 8 + wave;

  const float4* z4 = (const float4*)z;
  const float4* Wa4 = (const float4*)Wa;   // Wa: (1, 2*D)
  const float4* wr4 = (const float4*)wr;   // wr: (1,1,2,D)

  float4 zv = z4[(size_t)n * (DD / 4) + lane];
  float4 as = Wa4[lane];
  float4 ad = Wa4[(DD / 4) + lane];
  float4 w0 = wr4[lane];
  float4 w1 = wr4[(DD / 4) + lane];

  float d_as = zv.x * as.x + zv.y * as.y + zv.z * as.z + zv.w * as.w;
  float d_ad = zv.x * ad.x + zv.y * ad.y + zv.z * ad.z + zv.w * ad.w;
  float d_w0 = zv.x * w0.x + zv.y * w0.y + zv.z * w0.z + zv.w * w0.w;
  float d_w1 = zv.x * w1.x + zv.y * w1.y + zv.z * w1.z + zv.w * w1.w;

  d_as = wave_sum(d_as);
  d_ad = wave_sum(d_ad);
  d_w0 = wave_sum(d_w0);
  d_w1 = wave_sum(d_w1);

  if (lane == 0) {
    pack4[n] = make_float4(d_as, d_w0, d_w1, 0.0f);
    s_dst[n] = d_ad;
  }
}

// ---------------------------------------------------------------------------
// K3: attention + row/col convs + BN partials. One wave per node (lane == k).
// ---------------------------------------------------------------------------
__global__ __launch_bounds__(256) void gat_attn_kernel(
    const long long* __restrict__ nb, const float* __restrict__ z,
    const float4* __restrict__ pack4, const float* __restrict__ s_dst,
    const float* __restrict__ wc, const float* __restrict__ br,
    const float* __restrict__ bc, float* __restrict__ row_raw,
    float* __restrict__ col_raw, float4* __restrict__ partials) {
  const int lane = threadIdx.x & 31;
  const int wave = threadIdx.x >> 5;
  const int n = blockIdx.x * 8 + wave;

  __shared__ float lds[8][4];

  // --- attention logits + softmax across the wave (K == 32 lanes) ---
  int jj = (int)nb[(size_t)n * KK + lane];
  float4 p = pack4[jj];  // {s_src, t0, t1, -}
  float e = p.x + s_dst[n];
  e = (e > 0.0f) ? e : 0.01f * e;  // leaky_relu, slope 0.01
  float mx = wave_max(e);
  float ex = __expf(e - mx);
  float alpha = ex / wave_sum(ex);

  // --- row conv: row[kk] = a[kk]*t0[kk] + a[kk+1]*t1[kk+1] + br ---
  float r0 = alpha * p.y;
  float r1 = alpha * p.z;
  float rowv = r0 + __shfl_down(r1, 1, 32) + br[0];
  float rmask = (lane < (KK - 1)) ? 1.0f : 0.0f;
  if (lane < (KK - 1)) row_raw[(size_t)n * (KK - 1) + lane] = rowv;
  float sR = wave_sum(rowv * rmask);
  float qR = wave_sum(rowv * rowv * rmask);

  // --- col conv: col[d] = sum_k (wc[k]*a[k]) * z[nb[k]][d] + bc ---
  float coef = alpha * wc[lane];
  const float4* z4 = (const float4*)z;
  float4 acc = make_float4(0.f, 0.f, 0.f, 0.f);
#pragma unroll 4
  for (int k = 0; k < KK; ++k) {
    float ck = __shfl(coef, k, 32);
    int jk = __shfl(jj, k, 32);
    if (k + 1 < KK) {  // prefetch next gathered z row (global_prefetch_b8)
      int jn = __shfl(jj, k + 1, 32);
      __builtin_prefetch(&z4[(size_t)jn * (DD / 4) + lane], 0, 0);
    }
    float4 zv = z4[(size_t)jk * (DD / 4) + lane];
    acc.x = fmaf(ck, zv.x, acc.x);
    acc.y = fmaf(ck, zv.y, acc.y);
    acc.z = fmaf(ck, zv.z, acc.z);
    acc.w = fmaf(ck, zv.w, acc.w);
  }
  float b = bc[0];
  acc.x += b; acc.y += b; acc.z += b; acc.w += b;
  ((float4*)col_raw)[(size_t)n * (DD / 4) + lane] = acc;
  float sC = wave_sum(acc.x + acc.y + acc.z + acc.w);
  float qC = wave_sum(acc.x * acc.x + acc.y * acc.y + acc.z * acc.z +
                      acc.w * acc.w);

  // --- deterministic per-block BN partials ---
  if (lane == 0) {
    lds[wave][0] = sR; lds[wave][1] = qR; lds[wave][2] = sC; lds[wave][3] = qC;
  }
  __syncthreads();
  if (threadIdx.x == 0) {
    float4 t = make_float4(0.f, 0.f, 0.f, 0.f);
#pragma unroll
    for (int w = 0; w < 8; ++w) {
      t.x += lds[w][0]; t.y += lds[w][1]; t.z += lds[w][2]; t.w += lds[w][3];
    }
    partials[blockIdx.x] = t;
  }
}

// ---------------------------------------------------------------------------
// K4: reduce 3750 partials (fixed order -> deterministic), emit BN params.
// stats = { meanR, gr*invstdR, betar, meanC, gc*invstdC, betac }
// ---------------------------------------------------------------------------
__global__ __launch_bounds__(256) void gat_stats_kernel(
    const float4* __restrict__ partials, int nparts,
    const float* __restrict__ gr, const float* __restrict__ betar,
    const float* __restrict__ gc, const float* __restrict__ betac,
    float* __restrict__ stats) {
  __shared__ float4 lds[256];
  float4 t = make_float4(0.f, 0.f, 0.f, 0.f);
  for (int i = threadIdx.x; i < nparts; i += 256) {
    float4 q = partials[i];
    t.x += q.x; t.y += q.y; t.z += q.z; t.w += q.w;
  }
  lds[threadIdx.x] = t;
  __syncthreads();
  for (int s = 128; s >= 1; s >>= 1) {
    if (threadIdx.x < s) {
      float4 a = lds[threadIdx.x], c = lds[threadIdx.x + s];
      a.x += c.x; a.y += c.y; a.z += c.z; a.w += c.w;
      lds[threadIdx.x] = a;
    }
    __syncthreads();
  }
  if (threadIdx.x == 0) {
    float4 s0 = lds[0];
    const float cntR = (float)NN * (float)(KK - 1);
    const float cntC = (float)NN * (float)DD;
    float mR = s0.x / cntR;
    float vR = s0.y / cntR - mR * mR;
    float mC = s0.z / cntC;
    float vC = s0.w / cntC - mC * mC;
    stats[0] = mR;
    stats[1] = gr[0] * rsqrtf(vR + 1e-5f);
    stats[2] = betar[0];
    stats[3] = mC;
    stats[4] = gc[0] * rsqrtf(vC + 1e-5f);
    stats[5] = betac[0];
  }
}

// ---------------------------------------------------------------------------
// K5: normalize + relu + concat: out[n,0:31]=row, out[n,31:159]=col.
// ---------------------------------------------------------------------------
__global__ void gat_final_kernel(const float* __restrict__ row_raw,
                                 const float* __restrict__ col_raw,
                                 const float* __restrict__ stats,
                                 float* __restrict__ out) {
  const int n = blockIdx.x;
  const int r = threadIdx.x;
  if (r >= (KK - 1) + DD) return;
  float y;
  if (r < (KK - 1)) {
    float x = row_raw[(size_t)n * (KK - 1) + r];
    y = stats[1] * (x - stats[0]) + stats[2];
  } else {
    float x = col_raw[(size_t)n * DD + (r - (KK - 1))];
    y = stats[4] * (x - stats[3]) + stats[5];
  }
  out[(size_t)n * ((KK - 1) + DD) + r] = fmaxf(y, 0.0f);
}

// ---------------------------------------------------------------------------
extern "C" void kernel_launch(void* const* d_in, const int* in_sizes, int n_in,
                              void* d_out, int out_size, void* d_ws,
                              size_t ws_size, hipStream_t stream) {
  const float* h     = (const float*)d_in[0];
  const float* Wfc   = (const float*)d_in[1];
  const float* Wa    = (const float*)d_in[2];
  const float* wr    = (const float*)d_in[3];
  const float* br    = (const float*)d_in[4];
  const float* gr    = (const float*)d_in[5];
  const float* betar = (const float*)d_in[6];
  const float* wc    = (const float*)d_in[7];
  const float* bc    = (const float*)d_in[8];
  const float* gc    = (const float*)d_in[9];
  const float* betac = (const float*)d_in[10];
  const long long* nb = (const long long*)d_in[11];
  float* out = (float*)d_out;

  // workspace layout (floats); every region fully rewritten each call
  float* ws = (float*)d_ws;
  float*  z        = ws;                               // N*D   = 3,840,000
  float4* pack4    = (float4*)(z + (size_t)NN * DD);   // N*4
  float*  sdst     = (float*)pack4 + (size_t)NN * 4;   // N
  float*  row_raw  = sdst + NN;                        // N*31
  float*  col_raw  = row_raw + (size_t)NN * (KK - 1);  // N*D
  float4* partials = (float4*)(col_raw + (size_t)NN * DD);  // 3750*4
  float*  stats    = (float*)partials + 3750 * 4;      // 6

  gat_gemm_kernel<<<NN / 16, 256, 0, stream>>>(h, Wfc, z);
  gat_dots_kernel<<<NN / 8, 256, 0, stream>>>(z, Wa, wr, pack4, sdst);
  gat_attn_kernel<<<NN / 8, 256, 0, stream>>>(nb, z, pack4, sdst, wc, br, bc,
                                              row_raw, col_raw, partials);
  gat_stats_kernel<<<1, 256, 0, stream>>>(partials, NN / 8, gr, betar, gc,
                                          betac, stats);
  gat_final_kernel<<<NN, 160, 0, stream>>>(row_raw, col_raw, stats, out);
}